// GaussianVectorQuantizer_58772332478641
// MI455X (gfx1250) — compile-verified
//
#include <hip/hip_runtime.h>
#include <hip/hip_bf16.h>

typedef float v2f __attribute__((ext_vector_type(2)));
typedef float v8f __attribute__((ext_vector_type(8)));

#define NROWS   38912      // B*NPTS = 2048*19
#define NDIM    256
#define BOOKN   2048
#define GEPS    1e-10f
#define LDSPAD  260        // 256 + 4 pad -> conflict-free LDS rows

// ---------------------------------------------------------------- transpose
__global__ __launch_bounds__(256) void bookT_kernel(const float* __restrict__ book,
                                                    float* __restrict__ bookT) {
    int j = blockIdx.x;          // book row (0..2047)
    int d = threadIdx.x;         // dim (0..255)
    bookT[(size_t)d * BOOKN + j] = book[(size_t)j * NDIM + d];
}

// ---------------------------------------------------------------- row norms
__global__ __launch_bounds__(256) void norms_kernel(const float* __restrict__ ze,
                                                    const float* __restrict__ book,
                                                    float* __restrict__ zn,
                                                    float* __restrict__ bn) {
    int wave = threadIdx.x >> 5, lane = threadIdx.x & 31;
    int row = blockIdx.x * 8 + wave;     // 0..40959
    const float* src;
    float* dst;
    if (row < NROWS) { src = ze + (size_t)row * NDIM;             dst = zn + row; }
    else             { src = book + (size_t)(row - NROWS) * NDIM; dst = bn + (row - NROWS); }
    float s = 0.f;
    #pragma unroll
    for (int i = 0; i < 8; ++i) { float v = src[lane + i * 32]; s += v * v; }
    #pragma unroll
    for (int off = 16; off; off >>= 1) s += __shfl_xor(s, off, 32);
    if (lane == 0) *dst = s;
}

// ---------------------------------------------------------------- precision_q
__global__ void prec_kernel(const float* __restrict__ log_param_q, float* __restrict__ prec_out) {
    prec_out[0] = 0.5f / fmaxf(__expf(log_param_q[0]), GEPS);
}

// ---------------------------------------------------------------- GEMM1: logits
// logits[n,k] = -prec * (||ze_n||^2 + ||book_k||^2 - 2 * ze_n . book_k)
// Each wave handles two book-tiles (jt, jt+64) per iteration: one shared A
// fragment feeds two independent WMMA accumulation chains.
__global__ __launch_bounds__(256) void logits_kernel(const float* __restrict__ ze,
                                                     const float* __restrict__ book,
                                                     const float* __restrict__ bn,
                                                     const float* __restrict__ zn,
                                                     const float* __restrict__ log_param_q,
                                                     float* __restrict__ logits) {
    __shared__ float As[16 * LDSPAD];
    __shared__ float Zn[16];
    const int m0 = blockIdx.x * 16;
    const int t = threadIdx.x;

    // stage 16x256 ze tile (coalesced), padded rows
    #pragma unroll
    for (int i = 0; i < 16; ++i) {
        int e = t + i * 256;
        As[(e >> 8) * LDSPAD + (e & 255)] = ze[(size_t)m0 * NDIM + e];
    }
    if (t < 16) Zn[t] = zn[m0 + t];
    __syncthreads();

    const float prec = 0.5f / fmaxf(__expf(log_param_q[0]), GEPS);
    const int wave = t >> 5, lane = t & 31;
    const int col = lane & 15;               // N (and A-row M) index for this lane
    const int sel = (lane >> 4) << 1;        // K sub-select: 0 for lanes 0-15, 2 for 16-31
    const int rsel = (lane >> 4) << 3;       // C/D row offset: +8 for upper half lanes
    const float* Arow = &As[col * LDSPAD];

    for (int jt = wave; jt < 64; jt += 8) {
        const int j0 = jt * 16;              // first tile
        const int j1 = j0 + 1024;            // second tile (jt + 64)
        const float* bp0 = book + (size_t)(j0 + col) * NDIM + sel;
        const float* bp1 = book + (size_t)(j1 + col) * NDIM + sel;
        v8f acc0 = {}, acc1 = {};
        #pragma unroll 8
        for (int kk = 0; kk < 64; ++kk) {
            v2f a  = *(const v2f*)(Arow + kk * 4 + sel);
            v2f b0 = *(const v2f*)(bp0 + kk * 4);
            v2f b1 = *(const v2f*)(bp1 + kk * 4);
            acc0 = __builtin_amdgcn_wmma_f32_16x16x4_f32(false, a, false, b0,
                                                         (short)0, acc0, false, false);
            acc1 = __builtin_amdgcn_wmma_f32_16x16x4_f32(false, a, false, b1,
                                                         (short)0, acc1, false, false);
        }
        const float bnc0 = bn[j0 + col];
        const float bnc1 = bn[j1 + col];
        #pragma unroll
        for (int v = 0; v < 8; ++v) {
            int r = m0 + v + rsel;
            float zr = Zn[v + rsel];
            float d0 = zr + bnc0 - 2.0f * acc0[v];
            float d1 = zr + bnc1 - 2.0f * acc1[v];
            logits[(size_t)r * BOOKN + j0 + col] = -d0 * prec;
            logits[(size_t)r * BOOKN + j1 + col] = -d1 * prec;
        }
    }
}

// ---------------------------------------------------------------- softmax row stats
__global__ __launch_bounds__(256) void stats_kernel(const float* __restrict__ logits,
                                                    const float* __restrict__ gum,
                                                    const float* __restrict__ temperature,
                                                    float* __restrict__ rowM,
                                                    float* __restrict__ rowS) {
    const int wave = threadIdx.x >> 5, lane = threadIdx.x & 31;
    const int row = blockIdx.x * 8 + wave;
    const float invT = 1.0f / temperature[0];
    const float* lp = logits + (size_t)row * BOOKN;
    const float* gp = gum + (size_t)row * BOOKN;
    float m = -3.402823466e38f, s = 0.f;
    for (int i = 0; i < 64; ++i) {
        int k = lane + i * 32;
        float g = -__logf(-__logf(gp[k] + GEPS) + GEPS);
        float z = (lp[k] + g) * invT;
        float mn = fmaxf(m, z);
        s = s * __expf(m - mn) + __expf(z - mn);
        m = mn;
    }
    #pragma unroll
    for (int off = 16; off; off >>= 1) {
        float m2 = __shfl_xor(m, off, 32);
        float s2 = __shfl_xor(s, off, 32);
        float mn = fmaxf(m, m2);
        s = s * __expf(m - mn) + s2 * __expf(m2 - mn);
        m = mn;
    }
    if (lane == 0) { rowM[row] = m; rowS[row] = s; }
}

// ---------------------------------------------------------------- GEMM2: zq = softmax @ book
__global__ __launch_bounds__(256) void zq_kernel(const float* __restrict__ logits,
                                                 const float* __restrict__ gum,
                                                 const float* __restrict__ bookT,
                                                 const float* __restrict__ temperature,
                                                 const float* __restrict__ rowM,
                                                 const float* __restrict__ rowS,
                                                 float* __restrict__ zq) {
    __shared__ float Es[16 * LDSPAD];
    __shared__ float Ms[16], iSs[16];
    const int m0 = blockIdx.x * 16;
    const int t = threadIdx.x;
    if (t < 16) { Ms[t] = rowM[m0 + t]; iSs[t] = 1.0f / rowS[m0 + t]; }
    const float invT = 1.0f / temperature[0];
    const int wave = t >> 5, lane = t & 31;
    const int col = lane & 15;
    const int sel = (lane >> 4) << 1;
    const int rsel = (lane >> 4) << 3;
    const int d0 = wave * 32;                 // this wave's 32 output dims
    const float* b0base = bookT + (size_t)(d0 + col) * BOOKN + sel;
    const float* b1base = bookT + (size_t)(d0 + 16 + col) * BOOKN + sel;
    const float* Arow = &Es[col * LDSPAD];
    v8f acc0 = {}, acc1 = {};
    __syncthreads();

    for (int kc = 0; kc < BOOKN; kc += 256) {
        // stage E tile: encodings for rows m0..m0+15, cols kc..kc+255
        #pragma unroll
        for (int i = 0; i < 16; ++i) {
            int e = t + i * 256;
            int r = e >> 8, c = e & 255;
            size_t idx = (size_t)(m0 + r) * BOOKN + kc + c;
            float g = -__logf(-__logf(gum[idx] + GEPS) + GEPS);
            float z = (logits[idx] + g) * invT;
            Es[r * LDSPAD + c] = __expf(z - Ms[r]) * iSs[r];
        }
        __syncthreads();
        #pragma unroll 8
        for (int kk = 0; kk < 64; ++kk) {
            v2f a  = *(const v2f*)(Arow + kk * 4 + sel);
            v2f b0 = *(const v2f*)(b0base + kc + kk * 4);
            v2f b1 = *(const v2f*)(b1base + kc + kk * 4);
            acc0 = __builtin_amdgcn_wmma_f32_16x16x4_f32(false, a, false, b0,
                                                         (short)0, acc0, false, false);
            acc1 = __builtin_amdgcn_wmma_f32_16x16x4_f32(false, a, false, b1,
                                                         (short)0, acc1, false, false);
        }
        __syncthreads();
    }
    #pragma unroll
    for (int v = 0; v < 8; ++v) {
        int r = m0 + v + rsel;
        zq[(size_t)r * NDIM + d0 + col]      = acc0[v];
        zq[(size_t)r * NDIM + d0 + 16 + col] = acc1[v];
    }
}

// ---------------------------------------------------------------- launcher
extern "C" void kernel_launch(void* const* d_in, const int* in_sizes, int n_in,
                              void* d_out, int out_size, void* d_ws, size_t ws_size,
                              hipStream_t stream) {
    const float* ze          = (const float*)d_in[0];   // [38912,256]
    const float* temperature = (const float*)d_in[1];   // [1]
    const float* gum         = (const float*)d_in[2];   // [38912,2048]
    const float* book        = (const float*)d_in[3];   // [2048,256]
    const float* log_param_q = (const float*)d_in[4];   // [1]

    float* out      = (float*)d_out;
    float* zq       = out;                            // 9,961,472 floats
    float* prec_out = out + (size_t)NROWS * NDIM;     // 1 float
    float* logits   = prec_out + 1;                   // 79,691,776 floats

    float* ws    = (float*)d_ws;
    float* bookT = ws;                 // 524288 floats (256 x 2048)
    float* bn    = ws + 524288;        // 2048
    float* zn    = ws + 526336;        // 38912
    float* rowM  = ws + 565248;        // 38912
    float* rowS  = ws + 604160;        // 38912  (total ~2.46 MB)

    bookT_kernel <<<BOOKN, 256, 0, stream>>>(book, bookT);
    norms_kernel <<<(NROWS + BOOKN) / 8, 256, 0, stream>>>(ze, book, zn, bn);
    prec_kernel  <<<1, 1, 0, stream>>>(log_param_q, prec_out);
    logits_kernel<<<NROWS / 16, 256, 0, stream>>>(ze, book, bn, zn, log_param_q, logits);
    stats_kernel <<<NROWS / 8, 256, 0, stream>>>(logits, gum, temperature, rowM, rowS);
    zq_kernel    <<<NROWS / 16, 256, 0, stream>>>(logits, gum, bookT, temperature,
                                                  rowM, rowS, zq);
}